// GATModel_79328045957403
// MI455X (gfx1250) — compile-verified
//
#include <hip/hip_runtime.h>

typedef float v2f __attribute__((ext_vector_type(2)));
typedef float v8f __attribute__((ext_vector_type(8)));

#define F0 128
#define U0 64
#define H0 8
#define A0 8
#define U1 40

// ---------------------------------------------------------------------------
// Monotonic uint key for float atomic max (total order, handles negatives)
// ---------------------------------------------------------------------------
__device__ __forceinline__ unsigned fkey(float f) {
    unsigned b = __float_as_uint(f);
    return (b & 0x80000000u) ? ~b : (b | 0x80000000u);
}
__device__ __forceinline__ float fkey_dec(unsigned k) {
    return __uint_as_float((k & 0x80000000u) ? (k ^ 0x80000000u) : ~k);
}

__device__ __forceinline__ void edge_rc(const int* __restrict__ ei, int E, int e,
                                        int& r, int& c) {
    if (e < E) { r = ei[e]; c = ei[E + e]; }
    else       { r = e - E; c = e - E; }       // appended self loops
}

// ---------------------------------------------------------------------------
// Layer-0 projections: Q0=relu(X@wq0+bq0), K0=relu(X@wk0+bk0), V0=X@wv0
// One wave computes one 16x16 tile via v_wmma_f32_16x16x4_f32.
// grid.x = N/16 node tiles; grid.y = 12 (mat*4 + coltile)
// ---------------------------------------------------------------------------
__global__ __launch_bounds__(32) void proj0_kernel(
    const float* __restrict__ x,
    const float* __restrict__ wq, const float* __restrict__ bq,
    const float* __restrict__ wk, const float* __restrict__ bk,
    const float* __restrict__ wv,
    float* __restrict__ Q0, float* __restrict__ K0, float* __restrict__ V0)
{
    const int tile = blockIdx.x;
    const int mat  = blockIdx.y >> 2;   // 0=Q, 1=K, 2=V
    const int ct   = blockIdx.y & 3;    // column tile (U0/16 = 4)
    const int lane = threadIdx.x;
    const int hi   = lane >> 4;         // 0: K pair {0,1}; 1: K pair {2,3}
    const int lo   = lane & 15;

    const float* W   = (mat == 0) ? wq : (mat == 1) ? wk : wv;
    const float* B   = (mat == 0) ? bq : (mat == 1) ? bk : nullptr;
    float*       OUT = (mat == 0) ? Q0 : (mat == 1) ? K0 : V0;

    const int m = tile * 16 + lo;       // A-matrix row (per ISA A layout)
    const int n = ct * 16 + lo;         // B-matrix col (per ISA B/D layout)

    v8f acc = {};
    for (int k = 0; k < F0; k += 4) {
        const int kb = k + 2 * hi;
        v2f a, b;
        a.x = x[(size_t)m * F0 + kb];
        a.y = x[(size_t)m * F0 + kb + 1];
        b.x = W[(size_t)kb * U0 + n];
        b.y = W[(size_t)(kb + 1) * U0 + n];
        acc = __builtin_amdgcn_wmma_f32_16x16x4_f32(
                  false, a, false, b, (short)0, acc, false, false);
    }

    const float bias = (B != nullptr) ? B[n] : 0.0f;
    #pragma unroll
    for (int v = 0; v < 8; ++v) {
        const int row = tile * 16 + v + 8 * hi;   // D layout: M = v + 8*hi
        float val = acc[v] + bias;
        if (mat != 2) val = fmaxf(val, 0.0f);
        OUT[(size_t)row * U0 + n] = val;
    }
}

// ---------------------------------------------------------------------------
// Layer-1 V projection: v1 = h0 @ wv1  (64 -> 40, 3 masked col tiles)
// ---------------------------------------------------------------------------
__global__ __launch_bounds__(32) void proj1v_kernel(
    const float* __restrict__ h0, const float* __restrict__ wv1,
    float* __restrict__ v1)
{
    const int tile = blockIdx.x;
    const int ct   = blockIdx.y;        // 0..2 (48 cols, mask >= 40)
    const int lane = threadIdx.x;
    const int hi   = lane >> 4;
    const int lo   = lane & 15;

    const int m = tile * 16 + lo;
    const int n = ct * 16 + lo;
    const bool valid = (n < U1);

    v8f acc = {};
    for (int k = 0; k < U0; k += 4) {
        const int kb = k + 2 * hi;
        v2f a, b;
        a.x = h0[(size_t)m * U0 + kb];
        a.y = h0[(size_t)m * U0 + kb + 1];
        b.x = valid ? wv1[(size_t)kb * U1 + n] : 0.0f;
        b.y = valid ? wv1[(size_t)(kb + 1) * U1 + n] : 0.0f;
        acc = __builtin_amdgcn_wmma_f32_16x16x4_f32(
                  false, a, false, b, (short)0, acc, false, false);
    }
    if (valid) {
        #pragma unroll
        for (int v = 0; v < 8; ++v) {
            const int row = tile * 16 + v + 8 * hi;
            v1[(size_t)row * U1 + n] = acc[v];
        }
    }
}

// ---------------------------------------------------------------------------
// Layer-1 scalar attention heads: q1 = relu(h0@wq1+bq1), k1 = relu(h0@wk1+bk1)
// ---------------------------------------------------------------------------
__global__ void qk1_kernel(const float* __restrict__ h0,
                           const float* __restrict__ wq1, const float* __restrict__ bq1,
                           const float* __restrict__ wk1, const float* __restrict__ bk1,
                           float* __restrict__ q1, float* __restrict__ k1, int n)
{
    const int i = blockIdx.x * blockDim.x + threadIdx.x;
    if (i >= n) return;
    const float* hr = h0 + (size_t)i * U0;
    float q = bq1[0], k = bk1[0];
    #pragma unroll 8
    for (int j = 0; j < U0; ++j) {
        const float h = hr[j];
        q += h * wq1[j];
        k += h * wk1[j];
    }
    q1[i] = fmaxf(q, 0.0f);
    k1[i] = fmaxf(k, 0.0f);
}

// ---------------------------------------------------------------------------
// Layer-0 edge passes: one thread per (edge, head)
// ---------------------------------------------------------------------------
__global__ void edge_max0_kernel(const int* __restrict__ ei, int E, int Etot,
                                 const float* __restrict__ Q0,
                                 const float* __restrict__ K0,
                                 unsigned* __restrict__ max0key)
{
    const int t = blockIdx.x * blockDim.x + threadIdx.x;
    if (t >= Etot * H0) return;
    const int e = t >> 3, h = t & 7;
    int r, c; edge_rc(ei, E, e, r, c);
    const float* q = Q0 + (size_t)r * U0 + h * A0;
    const float* k = K0 + (size_t)c * U0 + h * A0;
    float s = 0.0f;
    #pragma unroll
    for (int a = 0; a < A0; ++a) s += q[a] * k[a];
    s *= 0.35355339059327373f;   // 1/sqrt(8)
    atomicMax(&max0key[r * H0 + h], fkey(s));
}

__global__ void edge_scatter0_kernel(const int* __restrict__ ei, int E, int Etot,
                                     const float* __restrict__ Q0,
                                     const float* __restrict__ K0,
                                     const float* __restrict__ V0,
                                     const unsigned* __restrict__ max0key,
                                     float* __restrict__ denom0,
                                     float* __restrict__ accum0)
{
    const int t = blockIdx.x * blockDim.x + threadIdx.x;
    if (t >= Etot * H0) return;
    const int e = t >> 3, h = t & 7;
    int r, c; edge_rc(ei, E, e, r, c);
    const float* q = Q0 + (size_t)r * U0 + h * A0;
    const float* k = K0 + (size_t)c * U0 + h * A0;
    float s = 0.0f;
    #pragma unroll
    for (int a = 0; a < A0; ++a) s += q[a] * k[a];
    s *= 0.35355339059327373f;
    const float m  = fkey_dec(max0key[r * H0 + h]);
    const float ev = __expf(s - m);
    atomicAdd(&denom0[r * H0 + h], ev);
    const float* vv  = V0 + (size_t)c * U0 + h * A0;
    float*       acc = accum0 + (size_t)r * U0 + h * A0;
    #pragma unroll
    for (int d = 0; d < A0; ++d) atomicAdd(&acc[d], ev * vv[d]);
}

// h0 = relu(accum0/denom0 + b0)
__global__ void finalize0_kernel(const float* __restrict__ accum0,
                                 const float* __restrict__ denom0,
                                 const float* __restrict__ b0,
                                 float* __restrict__ h0, int n)
{
    const int t = blockIdx.x * blockDim.x + threadIdx.x;
    if (t >= n * U0) return;
    const int nn = t >> 6, cc = t & 63;
    const float val = accum0[t] / denom0[nn * H0 + (cc >> 3)] + b0[cc];
    h0[t] = fmaxf(val, 0.0f);
}

// ---------------------------------------------------------------------------
// Layer-1 edge passes (single head, a_units = 1)
// ---------------------------------------------------------------------------
__global__ void edge_max1_kernel(const int* __restrict__ ei, int E, int Etot,
                                 const float* __restrict__ q1,
                                 const float* __restrict__ k1,
                                 unsigned* __restrict__ max1key)
{
    const int e = blockIdx.x * blockDim.x + threadIdx.x;
    if (e >= Etot) return;
    int r, c; edge_rc(ei, E, e, r, c);
    atomicMax(&max1key[r], fkey(q1[r] * k1[c]));
}

// one thread per (edge, j∈0..7); each covers d = j, j+8, ... < 40
__global__ void edge_scatter1_kernel(const int* __restrict__ ei, int E, int Etot,
                                     const float* __restrict__ q1,
                                     const float* __restrict__ k1,
                                     const float* __restrict__ v1,
                                     const unsigned* __restrict__ max1key,
                                     float* __restrict__ denom1,
                                     float* __restrict__ accum1)
{
    const int t = blockIdx.x * blockDim.x + threadIdx.x;
    if (t >= Etot * 8) return;
    const int e = t >> 3, j = t & 7;
    int r, c; edge_rc(ei, E, e, r, c);
    const float s  = q1[r] * k1[c];
    const float m  = fkey_dec(max1key[r]);
    const float ev = __expf(s - m);
    if (j == 0) atomicAdd(&denom1[r], ev);
    const float* vv  = v1 + (size_t)c * U1;
    float*       acc = accum1 + (size_t)r * U1;
    for (int d = j; d < U1; d += 8) atomicAdd(&acc[d], ev * vv[d]);
}

// out = accum1/denom1 + b1   (identity activation)
__global__ void finalize_out_kernel(const float* __restrict__ accum1,
                                    const float* __restrict__ denom1,
                                    const float* __restrict__ b1,
                                    float* __restrict__ out, int n)
{
    const int t = blockIdx.x * blockDim.x + threadIdx.x;
    if (t >= n * U1) return;
    out[t] = accum1[t] / denom1[t / U1] + b1[t % U1];
}

// ---------------------------------------------------------------------------
extern "C" void kernel_launch(void* const* d_in, const int* in_sizes, int n_in,
                              void* d_out, int out_size, void* d_ws, size_t ws_size,
                              hipStream_t stream)
{
    const float* x    = (const float*)d_in[0];
    const int*   ei   = (const int*)  d_in[1];
    const float* wq0  = (const float*)d_in[2];
    const float* bq0  = (const float*)d_in[3];
    const float* wk0  = (const float*)d_in[4];
    const float* bk0  = (const float*)d_in[5];
    const float* wv0  = (const float*)d_in[6];
    const float* b0   = (const float*)d_in[7];
    const float* wq1  = (const float*)d_in[8];
    const float* bq1  = (const float*)d_in[9];
    const float* wk1  = (const float*)d_in[10];
    const float* bk1  = (const float*)d_in[11];
    const float* wv1  = (const float*)d_in[12];
    const float* b1   = (const float*)d_in[13];
    float* out = (float*)d_out;

    const int N    = in_sizes[0] / F0;   // 50000
    const int E    = in_sizes[1] / 2;    // 800000
    const int Etot = E + N;              // + self loops

    // Workspace layout (floats), with region reuse:
    //  R0: Q0 (N*64)            -> later h0
    //  R1: K0 (N*64)            -> later v1 (N*40)
    //  R2: V0 (N*64)            -> later accum1 (N*40)
    //  R3: max0key (uint N*8)   -> later q1(N), k1(N)
    //  R4: denom0 (N*8)         -> later max1key(uint N), denom1(N)
    //  R5: accum0 (N*64)
    float* ws = (float*)d_ws;
    float* R0 = ws;
    float* R1 = R0 + (size_t)N * U0;
    float* R2 = R1 + (size_t)N * U0;
    float* R3 = R2 + (size_t)N * U0;
    float* R4 = R3 + (size_t)N * H0;
    float* R5 = R4 + (size_t)N * H0;

    const int TB = 256;

    // ---- Layer 0 ----
    proj0_kernel<<<dim3(N / 16, 12), 32, 0, stream>>>(
        x, wq0, bq0, wk0, bk0, wv0, R0, R1, R2);

    // zero max0key + denom0 + accum0 (contiguous: 80*N floats at R3)
    hipMemsetAsync(R3, 0, (size_t)N * (H0 + H0 + U0) * sizeof(float), stream);

    edge_max0_kernel<<<(Etot * H0 + TB - 1) / TB, TB, 0, stream>>>(
        ei, E, Etot, R0, R1, (unsigned*)R3);
    edge_scatter0_kernel<<<(Etot * H0 + TB - 1) / TB, TB, 0, stream>>>(
        ei, E, Etot, R0, R1, R2, (const unsigned*)R3, R4, R5);
    finalize0_kernel<<<(N * U0 + TB - 1) / TB, TB, 0, stream>>>(
        R5, R4, b0, R0 /* h0 overwrites Q0 */, N);

    // ---- Layer 1 ----
    float* q1 = R3;                 // reuse max0key region
    float* k1 = R3 + N;
    qk1_kernel<<<(N + TB - 1) / TB, TB, 0, stream>>>(
        R0, wq1, bq1, wk1, bk1, q1, k1, N);
    proj1v_kernel<<<dim3(N / 16, 3), 32, 0, stream>>>(
        R0, wv1, R1 /* v1 overwrites K0 */);

    unsigned* max1key = (unsigned*)R4;  // reuse denom0 region
    float*    denom1  = R4 + N;
    float*    accum1  = R2;             // reuse V0 region
    hipMemsetAsync(R4, 0, (size_t)2 * N * sizeof(float), stream);   // max1key+denom1
    hipMemsetAsync(R2, 0, (size_t)N * U1 * sizeof(float), stream);  // accum1

    edge_max1_kernel<<<(Etot + TB - 1) / TB, TB, 0, stream>>>(
        ei, E, Etot, q1, k1, max1key);
    edge_scatter1_kernel<<<(Etot * 8 + TB - 1) / TB, TB, 0, stream>>>(
        ei, E, Etot, q1, k1, R1, max1key, denom1, accum1);
    finalize_out_kernel<<<(N * U1 + TB - 1) / TB, TB, 0, stream>>>(
        accum1, denom1, b1, out, N);
}